// SkeletalDecBlock_44667659879002
// MI455X (gfx1250) — compile-verified
//
#include <hip/hip_runtime.h>
#include <hip/hip_bf16.h>

typedef float v2f __attribute__((ext_vector_type(2)));
typedef float v8f __attribute__((ext_vector_type(8)));

#define NE    28      // edges
#define NP    14      // pooled regions
#define NC    16      // channels per edge
#define TIN   4096
#define TOUT  8192
#define NB    16      // batch
#define KDIM  144     // 3 taps * 3 edges * 16 ch
#define TTILE 128     // output timesteps per block
#define NYT   (TTILE + 2)   // 130 staged upsampled samples (t0-1 .. t0+128)
#define TSTR  52            // row stride of time-major sY (48 rows padded to 52)

// K ordering: kk = k*48 + j*16 + ci   (k = tap, j = neighbor slot, ci = channel)
// sY layout:  sY[t][row],  row = j*16 + ci,  value needed for output col n, tap k
//             lives at  (nb + n + k)*TSTR + row

__global__ __launch_bounds__(128)
void skel_dec_kernel(const float* __restrict__ x,
                     const float* __restrict__ weight,
                     const float* __restrict__ bias,
                     float* __restrict__ out)
{
    __shared__ float sW[NC * KDIM];      // packed per-edge weights [co][kk]
    __shared__ float sY[NYT * TSTR];     // upsampled inputs, time-major [t][row]
    __shared__ float sB[NC];

    const int ntiles = TOUT / TTILE;     // 64
    int bid = blockIdx.x;
    int b   = bid / (NE * ntiles);
    int rem = bid % (NE * ntiles);
    int eo  = rem / ntiles;
    int t0  = (rem % ntiles) * TTILE;

    int tid = threadIdx.x;

    // ---- Phase 1: pack weights (zero outside skeleton => boundary edges exact)
    for (int idx = tid; idx < NC * KDIM; idx += 128) {
        int co = idx / KDIM, kk = idx % KDIM;
        int k = kk / 48, r = kk % 48, j = r / 16, ci = r % 16;
        int ei = eo - 1 + j;
        float w = 0.0f;
        if (ei >= 0 && ei < NE)
            w = weight[((size_t)(eo * NC + co) * (NE * NC) + (ei * NC + ci)) * 3 + k];
        sW[idx] = w;
    }
    if (tid < NC) sB[tid] = bias[eo * NC + tid];

    // ---- Phase 2: fused linear-2x upsample + unpool into time-major LDS
    for (int idx = tid; idx < 3 * NC * NYT; idx += 128) {
        int slot = idx / (NC * NYT);
        int r2   = idx % (NC * NYT);
        int ci   = r2 / NYT;
        int tt   = r2 % NYT;                 // tt fastest -> coalesced x reads
        int ei   = eo - 1 + slot;
        ei = ei < 0 ? 0 : (ei > NE - 1 ? NE - 1 : ei);
        int reg  = ei >> 1;                  // UNPOOL_IDX[e] = e/2
        int tg   = t0 - 1 + tt;
        float v = 0.0f;
        if (tg >= 0 && tg < TOUT) {          // zeros outside = conv padding
            const float* xr = x + ((size_t)b * (NP * NC) + reg * NC + ci) * (size_t)TIN;
            int th = tg >> 1;
            if ((tg & 1) == 0) {             // y[2t]   = 0.25 x[t-1] + 0.75 x[t]
                int lo = th - 1 < 0 ? 0 : th - 1;
                v = 0.25f * xr[lo] + 0.75f * xr[th];
            } else {                         // y[2t+1] = 0.75 x[t] + 0.25 x[t+1]
                int hi2 = th + 1 > TIN - 1 ? TIN - 1 : th + 1;
                v = 0.75f * xr[th] + 0.25f * xr[hi2];
            }
        }
        sY[tt * TSTR + slot * NC + ci] = v;
    }
    __syncthreads();

    // ---- Phase 3: WMMA GEMM  D[16x16] += A[16x4] * B[4x16], K = 144
    int wave = tid >> 5;
    int lane = tid & 31;
    int n    = lane & 15;
    bool hi  = lane >= 16;
    int koff = hi ? 2 : 0;

    // A VGPRs: lane<16 -> (K=4kb, 4kb+1); lane>=16 -> (K=4kb+2, 4kb+3); M = lane&15
    v2f a[36];
    {
        const float* wr = sW + (n * KDIM + koff);
#pragma unroll
        for (int kb = 0; kb < 36; ++kb)
            a[kb] = *(const v2f*)(wr + 4 * kb);
    }

    // per-lane base: column (wave*32 + n), row offset for lane half
    int laneBase = (wave * 32 + n) * TSTR + koff;
    v8f acc0 = {};
    v8f acc1 = {};

#pragma unroll
    for (int kb = 0; kb < 36; ++kb) {
        // compile-time: tap k and first row of this 4-K block
        const int Ck = ((4 * kb) / 48) * TSTR + ((4 * kb) % 48);
        int o = laneBase + Ck;
        v2f b0 = *(const v2f*)(sY + o);                 // rows r0+koff, r0+koff+1
        v2f b1 = *(const v2f*)(sY + o + 16 * TSTR);     // same, 16 columns later
        acc0 = __builtin_amdgcn_wmma_f32_16x16x4_f32(
            false, a[kb], false, b0, (short)0, acc0, false, false);
        acc1 = __builtin_amdgcn_wmma_f32_16x16x4_f32(
            false, a[kb], false, b1, (short)0, acc1, false, false);
    }

    // ---- Epilogue: bias + LeakyReLU(0.2) + store (C/D layout: VGPR v -> M=v / v+8)
    int tcol0 = t0 + wave * 32 + n;
    size_t obase = ((size_t)b * (NE * NC) + eo * NC) * (size_t)TOUT;
#pragma unroll
    for (int v = 0; v < 8; ++v) {
        int m = v + (hi ? 8 : 0);
        float bv = sB[m];
        float o0 = acc0[v] + bv;
        float o1 = acc1[v] + bv;
        o0 = o0 >= 0.0f ? o0 : 0.2f * o0;
        o1 = o1 >= 0.0f ? o1 : 0.2f * o1;
        out[obase + (size_t)m * TOUT + tcol0]      = o0;
        out[obase + (size_t)m * TOUT + tcol0 + 16] = o1;
    }
}

extern "C" void kernel_launch(void* const* d_in, const int* in_sizes, int n_in,
                              void* d_out, int out_size, void* d_ws, size_t ws_size,
                              hipStream_t stream) {
    const float* x      = (const float*)d_in[0];   // [16, 224, 4096]
    const float* weight = (const float*)d_in[1];   // [448, 448, 3]
    const float* bias   = (const float*)d_in[2];   // [448]
    float* out          = (float*)d_out;           // [16, 448, 8192]
    (void)in_sizes; (void)n_in; (void)out_size; (void)d_ws; (void)ws_size;

    dim3 grid(NB * NE * (TOUT / TTILE));           // 16*28*64 = 28672 blocks
    skel_dec_kernel<<<grid, 128, 0, stream>>>(x, weight, bias, out);
}